// PreprocessHead_30468497998475
// MI455X (gfx1250) — compile-verified
//
#include <hip/hip_runtime.h>
#include <stdint.h>

// ---------------------------------------------------------------------------
// RPN proposal head for MI455X (gfx1250):
//   decode+score -> pack sort keys -> hybrid LDS/global bitonic sort (21
//   launches instead of 136; all j<=1024 spans run in LDS) -> greedy NMS with
//   the whole 12000-box candidate set resident in the 320KB WGP LDS.
//   Global->LDS staging uses gfx1250 async copies (ASYNCcnt path); NMS
//   suppression uses wave32 ballots (one 32-bit bitmap word == one wave).
//   No WMMA by design: the operator contains no inner products.
// ---------------------------------------------------------------------------

#define BATCH   8
#define NANCH   36864
#define NPAD    65536          // pow2 padding for bitonic sort (2^16)
#define NPRE    12000
#define NPOST   2000
#define WRD     (NPRE / 32)    // 375 bitmap words (NPRE is a multiple of 32)
#define NEGVAL  (-1.0e9f)
#define NMS_T   0.7f
#define SEG     2048           // LDS-resident bitonic segment (16KB of keys)

// ---------------------------------------------------------------------------
// Kernel 1: decode boxes, compute scores, pack sort keys.
// key = (orderable_score_bits << 32) | (0xFFFFFFFF - anchor_index)
// Descending u64 sort => descending score, ascending index on ties (matches
// jax.lax.top_k tie-break). Pad slots get key 0 (sorts below every real
// score, including NEG-masked ones: orderable(-1e9) = 0x319194D7 > 0).
// ---------------------------------------------------------------------------
__global__ void rpn_decode(const int* __restrict__ ih, const int* __restrict__ iw,
                           const float* __restrict__ anchors,
                           const float* __restrict__ loc,
                           const float* __restrict__ obj,
                           float* __restrict__ boxes,
                           unsigned long long* __restrict__ keys) {
  int t = blockIdx.x * blockDim.x + threadIdx.x;
  int b = t >> 16;             // NPAD == 2^16
  int i = t & (NPAD - 1);
  if (b >= BATCH) return;
  unsigned long long* kb = keys + ((size_t)b << 16);
  if (i >= NANCH) { kb[i] = 0ull; return; }

  float img_h = (float)ih[0];
  float img_w = (float)iw[0];

  const float* a = anchors + (size_t)i * 4;
  float aw = a[2] - a[0];
  float ah = a[3] - a[1];
  float cx = a[0] + 0.5f * aw;
  float cy = a[1] + 0.5f * ah;

  const float* l = loc + ((size_t)b * NANCH + i) * 4;
  float pcx = l[0] * aw + cx;
  float pcy = l[1] * ah + cy;
  float pw  = expf(l[2]) * aw;
  float ph  = expf(l[3]) * ah;

  float x1 = fminf(fmaxf(pcx - 0.5f * pw, 0.0f), img_w);
  float y1 = fminf(fmaxf(pcy - 0.5f * ph, 0.0f), img_h);
  float x2 = fminf(fmaxf(pcx + 0.5f * pw, 0.0f), img_w);
  float y2 = fminf(fmaxf(pcy + 0.5f * ph, 0.0f), img_h);

  const float* o = obj + ((size_t)b * NANCH + i) * 2;
  float s = 1.0f / (1.0f + expf(o[0] - o[1]));   // softmax(...)[1]
  bool valid = (x2 - x1 >= 16.0f) && (y2 - y1 >= 16.0f);
  if (!valid) s = NEGVAL;

  float4* bo = (float4*)(boxes + ((size_t)b * NANCH + i) * 4);
  *bo = make_float4(x1, y1, x2, y2);

  uint32_t ub  = __float_as_uint(s);
  uint32_t ord = (ub & 0x80000000u) ? ~ub : (ub | 0x80000000u); // orderable
  kb[i] = ((unsigned long long)ord << 32) | (unsigned long long)(0xFFFFFFFFu - (uint32_t)i);
}

// ---------------------------------------------------------------------------
// Bitonic helpers. Direction comes from the per-image index (gi & (NPAD-1)),
// so each image is sorted independently (descending).
// ---------------------------------------------------------------------------
__device__ __forceinline__ bool bitonic_up(int gi, int k) {
  return (((gi & (NPAD - 1)) & k) == 0);
}

// One global compare-exchange pass (only used for spans j >= SEG).
__global__ void rpn_bitonic_global(unsigned long long* __restrict__ keys, int k, int j) {
  int t = blockIdx.x * blockDim.x + threadIdx.x;
  int b = t >> 16;
  int i = t & (NPAD - 1);
  if (b >= BATCH) return;
  int l = i ^ j;
  if (l <= i) return;
  unsigned long long* kb = keys + ((size_t)b << 16);
  unsigned long long a = kb[i];
  unsigned long long c = kb[l];
  bool up  = bitonic_up(i, k);           // descending: up segments keep larger first
  bool swp = up ? (a < c) : (a > c);
  if (swp) { kb[i] = c; kb[l] = a; }
}

// LDS-resident passes over one SEG-sized segment. kstart==2 runs the full
// local triangle k=2..SEG (66 passes); otherwise runs the tail j=SEG/2..1 of
// a single merge step k=kfix (11 passes). Keys staged via async global->LDS.
__device__ __forceinline__ void rpn_bitonic_local_body(unsigned long long* __restrict__ keys,
                                                       int kstart, int kfix) {
  __shared__ unsigned long long skey[SEG];
  const int tid  = threadIdx.x;                 // 1024 threads, SEG/2 pairs
  const int base = blockIdx.x * SEG;            // never straddles an image

  // Stage 2 keys/thread (16B) with an async global->LDS b128 copy.
  {
    const unsigned long long* src = keys + base + 2 * tid;
    uint32_t ldsOff = (uint32_t)(uintptr_t)(const void*)&skey[2 * tid];
    asm volatile("global_load_async_to_lds_b128 %0, %1, off"
                 :: "v"(ldsOff), "v"(src) : "memory");
  }
  asm volatile("s_wait_asynccnt 0x0" ::: "memory");

  if (kstart == 2) {
    for (int k = 2; k <= SEG; k <<= 1) {
      for (int j = k >> 1; j >= 1; j >>= 1) {
        __syncthreads();
        int i = ((tid & ~(j - 1)) << 1) | (tid & (j - 1));
        int l = i | j;
        unsigned long long a = skey[i];
        unsigned long long c = skey[l];
        bool up  = bitonic_up(base + i, k);
        bool swp = up ? (a < c) : (a > c);
        if (swp) { skey[i] = c; skey[l] = a; }
      }
    }
  } else {
    int k = kfix;
    for (int j = SEG >> 1; j >= 1; j >>= 1) {
      __syncthreads();
      int i = ((tid & ~(j - 1)) << 1) | (tid & (j - 1));
      int l = i | j;
      unsigned long long a = skey[i];
      unsigned long long c = skey[l];
      bool up  = bitonic_up(base + i, k);
      bool swp = up ? (a < c) : (a > c);
      if (swp) { skey[i] = c; skey[l] = a; }
    }
  }
  __syncthreads();
  *(ulonglong2*)&keys[base + 2 * tid] =
      *(const ulonglong2*)&skey[2 * tid];
}

__global__ void __launch_bounds__(SEG / 2)
rpn_bitonic_local_sort(unsigned long long* __restrict__ keys) {
  rpn_bitonic_local_body(keys, 2, 0);
}

__global__ void __launch_bounds__(SEG / 2)
rpn_bitonic_local_merge(unsigned long long* __restrict__ keys, int k) {
  rpn_bitonic_local_body(keys, 0, k);
}

// ---------------------------------------------------------------------------
// Kernel 3: greedy NMS, one workgroup (32 waves) per image.
// Dynamic LDS (~195KB of the 320KB WGP pool):
//   float4 sbox[12000]  sorted top-12000 boxes (async-gathered)
//   u32    sup[375]     suppression bitmap (ballot-written, no atomics)
//   u32    vbit[375]    validity bitmap (score > NEG/2)
//   int    s_sel        broadcast slot
// ---------------------------------------------------------------------------
__global__ void __launch_bounds__(1024)
rpn_nms(const unsigned long long* __restrict__ keys,
        const float* __restrict__ boxes,
        float* __restrict__ outBoxes,
        float* __restrict__ outMask) {
  extern __shared__ char smem[];
  float4*   sbox  = (float4*)smem;
  uint32_t* sup   = (uint32_t*)(smem + (size_t)NPRE * 16);
  uint32_t* vbit  = sup + ((WRD + 1) & ~1);
  int*      s_sel = (int*)(vbit + ((WRD + 1) & ~1));

  const int tid = threadIdx.x;
  const int b   = blockIdx.x;
  const int T   = blockDim.x;
  const unsigned long long* kb = keys + ((size_t)b << 16);

  for (int w = tid; w < WRD; w += T) sup[w] = 0u;

  // Gather sorted top-NPRE boxes into LDS via per-lane async copies; validity
  // bits via wave32 ballot (ranks are lane-consecutive since T % 32 == 0).
  for (int r = tid; r < NPRE; r += T) {
    unsigned long long key = kb[r];
    uint32_t idx = 0xFFFFFFFFu - (uint32_t)(key & 0xFFFFFFFFull);
    const float* src = boxes + ((size_t)b * NANCH + idx) * 4;
    uint32_t ldsOff = (uint32_t)(uintptr_t)(const void*)&sbox[r];
    asm volatile("global_load_async_to_lds_b128 %0, %1, off"
                 :: "v"(ldsOff), "v"(src) : "memory");
    uint32_t ord = (uint32_t)(key >> 32);
    uint32_t ub  = (ord & 0x80000000u) ? (ord & 0x7FFFFFFFu) : ~ord; // un-orderable
    float s = __uint_as_float(ub);
    uint32_t m = (uint32_t)__ballot(s > 0.5f * NEGVAL);
    if ((tid & 31) == 0) vbit[r >> 5] = m;
  }
  asm volatile("s_wait_asynccnt 0x0" ::: "memory");
  __syncthreads();

  int c  = 0;   // proposals emitted
  int r0 = 0;   // sorted order => next argmax is the first unsuppressed rank
  while (c < NPOST) {
    if (tid == 0) {
      int r = r0;
      while (r < NPRE && ((sup[r >> 5] >> (r & 31)) & 1u)) ++r;
      *s_sel = r;
    }
    __syncthreads();
    int sel = *s_sel;
    if (sel >= NPRE) break;

    float4 bx = sbox[sel];
    if (tid == 0) {
      bool valid = (vbit[sel >> 5] >> (sel & 31)) & 1u;
      float* o = outBoxes + ((size_t)b * NPOST + c) * 4;
      o[0] = valid ? bx.x : 0.0f;
      o[1] = valid ? bx.y : 0.0f;
      o[2] = valid ? bx.z : 0.0f;
      o[3] = valid ? bx.w : 0.0f;
      outMask[(size_t)b * NPOST + c] = valid ? 1.0f : 0.0f;
    }

    // Suppress (IoU > thresh, self included). Each wave owns whole 32-bit
    // bitmap words -> ballot + plain OR, zero atomics.
    float a0 = (bx.z - bx.x) * (bx.w - bx.y);
    for (int jj = tid; jj < NPRE; jj += T) {
      float4 bj = sbox[jj];
      float ix1 = fmaxf(bx.x, bj.x);
      float iy1 = fmaxf(bx.y, bj.y);
      float ix2 = fminf(bx.z, bj.z);
      float iy2 = fminf(bx.w, bj.w);
      float inter = fmaxf(ix2 - ix1, 0.0f) * fmaxf(iy2 - iy1, 0.0f);
      float a1 = (bj.z - bj.x) * (bj.w - bj.y);
      float iou = inter / fmaxf(a0 + a1 - inter, 1e-9f);
      uint32_t m = (uint32_t)__ballot(iou > NMS_T);
      if ((tid & 31) == 0) sup[jj >> 5] |= m;
    }
    r0 = sel + 1;
    ++c;
    __syncthreads();
  }

  // Remaining slots: invalid (reference keeps emitting masked-out entries).
  for (int q = c + tid; q < NPOST; q += T) {
    float* o = outBoxes + ((size_t)b * NPOST + q) * 4;
    o[0] = 0.0f; o[1] = 0.0f; o[2] = 0.0f; o[3] = 0.0f;
    outMask[(size_t)b * NPOST + q] = 0.0f;
  }
}

// ---------------------------------------------------------------------------
extern "C" void kernel_launch(void* const* d_in, const int* in_sizes, int n_in,
                              void* d_out, int out_size, void* d_ws, size_t ws_size,
                              hipStream_t stream) {
  (void)in_sizes; (void)n_in; (void)out_size; (void)ws_size;
  const int*   ih      = (const int*)d_in[0];
  const int*   iw      = (const int*)d_in[1];
  const float* anchors = (const float*)d_in[2];
  const float* loc     = (const float*)d_in[3];
  const float* obj     = (const float*)d_in[4];

  float* boxes = (float*)d_ws;                                  // 4.72 MB
  unsigned long long* keys =
      (unsigned long long*)((char*)d_ws + (size_t)BATCH * NANCH * 4 * sizeof(float)); // 4 MB

  float* outBoxes = (float*)d_out;
  float* outMask  = outBoxes + (size_t)BATCH * NPOST * 4;

  const int total      = BATCH * NPAD;        // 524288
  const int segBlocks  = total / SEG;         // 256 blocks of 1024 threads

  rpn_decode<<<total / 256, 256, 0, stream>>>(ih, iw, anchors, loc, obj, boxes, keys);

  // Hybrid bitonic sort: 21 launches total.
  rpn_bitonic_local_sort<<<segBlocks, SEG / 2, 0, stream>>>(keys);     // k = 2..2048
  for (int k = SEG * 2; k <= NPAD; k <<= 1) {
    for (int j = k >> 1; j >= SEG; j >>= 1)
      rpn_bitonic_global<<<total / 256, 256, 0, stream>>>(keys, k, j); // j >= 2048
    rpn_bitonic_local_merge<<<segBlocks, SEG / 2, 0, stream>>>(keys, k); // j = 1024..1
  }

  const int smem_bytes = NPRE * 16 + 2 * (((WRD + 1) & ~1) * 4) + 16; // ~195 KB < 320 KB WGP LDS
  hipFuncSetAttribute(reinterpret_cast<const void*>(rpn_nms),
                      hipFuncAttributeMaxDynamicSharedMemorySize, smem_bytes);
  rpn_nms<<<BATCH, 1024, smem_bytes, stream>>>(keys, boxes, outBoxes, outMask);
}